// splineconv_66907000537299
// MI455X (gfx1250) — compile-verified
//
#include <hip/hip_runtime.h>

typedef float v2f __attribute__((ext_vector_type(2)));
typedef float v8f __attribute__((ext_vector_type(8)));

// ---------------------------------------------------------------------------
// Pack W[k][Cin][Ch] (k=0,1) and root[Cin][Ch] into Wcat[Cin][3*Ch]
// columns: [0,Ch) = knot0, [Ch,2Ch) = knot1, [2Ch,3Ch) = root
// ---------------------------------------------------------------------------
__global__ __launch_bounds__(256) void pack_w1_kernel(
    const float* __restrict__ W, const float* __restrict__ root,
    float* __restrict__ Wc) {
  int t = blockIdx.x * 256 + threadIdx.x;          // 128*96
  if (t >= 128 * 96) return;
  int i = t / 96, c = t % 96;
  float v;
  if (c < 32)        v = W[i * 32 + c];
  else if (c < 64)   v = W[128 * 32 + i * 32 + (c - 32)];
  else               v = root[i * 32 + (c - 64)];
  Wc[t] = v;
}

__global__ __launch_bounds__(256) void pack_w2_kernel(
    const float* __restrict__ W, const float* __restrict__ root,
    float* __restrict__ Wc) {
  int t = blockIdx.x * 256 + threadIdx.x;          // 32*192
  if (t >= 32 * 192) return;
  int i = t / 192, c = t % 192;
  float v;
  if (c < 64)        v = W[i * 64 + c];
  else if (c < 128)  v = W[32 * 64 + i * 64 + (c - 64)];
  else               v = root[i * 64 + (c - 128)];
  Wc[t] = v;
}

// ---------------------------------------------------------------------------
// GEMM via V_WMMA_F32_16X16X4_F32: Out[M,NCOLS] = A[M,KDIM] * W[KDIM,NCOLS]
// One wave per 16-row stripe, all NCOLS/16 column tiles; W staged in LDS.
// A frag (16x4): lane<16 -> K=0,1 ; lane>=16 -> K=2,3 (M = lane&15)
// B frag (4x16): same K split, N = lane&15
// D (16x16): vgpr r, lane L -> row r + (L<16?0:8), col L&15
// ---------------------------------------------------------------------------
template <int KDIM, int NCOLS>
__global__ __launch_bounds__(256) void gemm_wmma_kernel(
    const float* __restrict__ A, const float* __restrict__ W,
    float* __restrict__ Out, int M) {
  constexpr int NT = NCOLS / 16;
  __shared__ float sW[KDIM * NCOLS];
  for (int i = threadIdx.x; i < KDIM * NCOLS; i += 256) sW[i] = W[i];
  __syncthreads();

  const int wave  = threadIdx.x >> 5;
  const int lane  = threadIdx.x & 31;
  const int mtile = blockIdx.x * 8 + wave;
  if (mtile * 16 >= M) return;                      // uniform per-wave exit

  const int ln   = lane & 15;
  const int koff = (lane >> 4) << 1;                // 0 or 2
  int row = mtile * 16 + ln;
  if (row >= M) row = M - 1;                        // safe clamp (M%16==0 normally)
  const float* __restrict__ arow = A + (size_t)row * KDIM;

  v8f acc[NT] = {};
  for (int k0 = 0; k0 < KDIM; k0 += 4) {
    v2f a;
    a.x = arow[k0 + koff];
    a.y = arow[k0 + koff + 1];
    const float* b0 = &sW[(k0 + koff) * NCOLS + ln];
    const float* b1 = &sW[(k0 + koff + 1) * NCOLS + ln];
#pragma unroll
    for (int t = 0; t < NT; ++t) {
      v2f b;
      b.x = b0[t * 16];
      b.y = b1[t * 16];
      acc[t] = __builtin_amdgcn_wmma_f32_16x16x4_f32(
          false, a, false, b, (short)0, acc[t], false, false);
    }
  }

  const int rbase = mtile * 16 + ((lane >> 4) << 3);
#pragma unroll
  for (int t = 0; t < NT; ++t) {
#pragma unroll
    for (int r = 0; r < 8; ++r) {
      int orow = rbase + r;
      if (orow < M) Out[(size_t)orow * NCOLS + t * 16 + ln] = acc[t][r];
    }
  }
}

// ---------------------------------------------------------------------------
// Edge pass: lane = channel, wave(s) per edge. Gather two knot rows of T,
// lerp by pseudo, scatter with native f32 atomics. Optionally count degree.
// ---------------------------------------------------------------------------
template <int C, int STRIDE, bool DO_DEG>
__global__ __launch_bounds__(256) void edge_scatter_kernel(
    const int* __restrict__ src, const int* __restrict__ dst,
    const float* __restrict__ ps, const float* __restrict__ T,
    float* __restrict__ acc, float* __restrict__ deg, int E) {
  constexpr int LOG2C = (C == 32) ? 5 : 6;
  const unsigned t = blockIdx.x * 256u + threadIdx.x;
  const int e = (int)(t >> LOG2C);
  const int c = (int)(t & (C - 1));
  if (e >= E) return;
  const int s = src[e];
  const int d = dst[e];
  const float f = ps[e];
  const float* __restrict__ base = T + (size_t)s * STRIDE;
  const float a = base[c];          // knot 0
  const float b = base[C + c];      // knot 1
  const float m = fmaf(f, b - a, a);
  unsafeAtomicAdd(&acc[(size_t)d * C + c], m);
  if (DO_DEG && c == 0) unsafeAtomicAdd(&deg[d], 1.0f);
}

// ---------------------------------------------------------------------------
// Layer-1 epilogue: h = relu(acc/deg + root_part + b1), 1 thread / (node,ch)
// ---------------------------------------------------------------------------
__global__ __launch_bounds__(256) void node_relu_kernel(
    const float* __restrict__ acc, const float* __restrict__ deg,
    const float* __restrict__ T, const float* __restrict__ bias,
    float* __restrict__ h, int N) {
  const unsigned t = blockIdx.x * 256u + threadIdx.x;
  const int n = (int)(t >> 5);
  const int c = (int)(t & 31);
  if (n >= N) return;
  const float dg = fmaxf(deg[n], 1.0f);
  const float v = acc[(size_t)n * 32 + c] / dg
                + T[(size_t)n * 96 + 64 + c] + bias[c];
  h[(size_t)n * 32 + c] = fmaxf(v, 0.0f);
}

// ---------------------------------------------------------------------------
// Layer-2 epilogue + log_softmax over 64 channels: one wave32 per node,
// 2 channels per lane, max/sum via wave shuffles.
// ---------------------------------------------------------------------------
__global__ __launch_bounds__(256) void node_logsoftmax_kernel(
    const float* __restrict__ acc, const float* __restrict__ deg,
    const float* __restrict__ T, const float* __restrict__ bias,
    float* __restrict__ out, int N) {
  const unsigned gt = blockIdx.x * 256u + threadIdx.x;
  const int n = (int)(gt >> 5);
  const int lane = (int)(gt & 31);
  if (n >= N) return;
  const float dg = fmaxf(deg[n], 1.0f);
  float z0 = acc[(size_t)n * 64 + lane] / dg
           + T[(size_t)n * 192 + 128 + lane] + bias[lane];
  float z1 = acc[(size_t)n * 64 + 32 + lane] / dg
           + T[(size_t)n * 192 + 160 + lane] + bias[32 + lane];
  float mx = fmaxf(z0, z1);
#pragma unroll
  for (int o = 16; o > 0; o >>= 1) mx = fmaxf(mx, __shfl_xor(mx, o, 32));
  float ex = expf(z0 - mx) + expf(z1 - mx);
#pragma unroll
  for (int o = 16; o > 0; o >>= 1) ex += __shfl_xor(ex, o, 32);
  const float lse = mx + logf(ex);
  out[(size_t)n * 64 + lane]      = z0 - lse;
  out[(size_t)n * 64 + 32 + lane] = z1 - lse;
}

// ---------------------------------------------------------------------------
extern "C" void kernel_launch(void* const* d_in, const int* in_sizes, int n_in,
                              void* d_out, int out_size, void* d_ws, size_t ws_size,
                              hipStream_t stream) {
  const float* x  = (const float*)d_in[0];
  const int*   ei = (const int*)d_in[1];
  const float* ea = (const float*)d_in[2];
  const float* W1 = (const float*)d_in[3];
  const float* r1 = (const float*)d_in[4];
  const float* b1 = (const float*)d_in[5];
  const float* W2 = (const float*)d_in[6];
  const float* r2 = (const float*)d_in[7];
  const float* b2 = (const float*)d_in[8];

  const int N = in_sizes[0] / 128;
  const int E = in_sizes[2];
  const int* src = ei;
  const int* dst = ei + E;

  char* ws = (char*)d_ws;
  size_t off = 0;
  auto carve = [&](size_t nfloats) {
    size_t o = off;
    off += ((nfloats * sizeof(float)) + 255) & ~(size_t)255;
    return o;
  };
  float* Wc1 = (float*)(ws + carve(128 * 96));
  float* Wc2 = (float*)(ws + carve(32 * 192));
  float* T1  = (float*)(ws + carve((size_t)N * 96));
  float* ac1 = (float*)(ws + carve((size_t)N * 32));
  float* dg  = (float*)(ws + carve((size_t)N));
  float* h   = (float*)(ws + carve((size_t)N * 32));
  float* T2  = (float*)(ws + carve((size_t)N * 192));
  float* ac2 = (float*)(ws + carve((size_t)N * 64));

  hipMemsetAsync(ac1, 0, (size_t)N * 32 * sizeof(float), stream);
  hipMemsetAsync(dg,  0, (size_t)N * sizeof(float), stream);
  hipMemsetAsync(ac2, 0, (size_t)N * 64 * sizeof(float), stream);

  pack_w1_kernel<<<(128 * 96 + 255) / 256, 256, 0, stream>>>(W1, r1, Wc1);
  pack_w2_kernel<<<(32 * 192 + 255) / 256, 256, 0, stream>>>(W2, r2, Wc2);

  const int mtiles = (N + 15) / 16;
  const int gblocks = (mtiles + 7) / 8;

  gemm_wmma_kernel<128, 96><<<gblocks, 256, 0, stream>>>(x, Wc1, T1, N);

  {
    unsigned total = (unsigned)E * 32u;
    edge_scatter_kernel<32, 96, true>
        <<<(total + 255) / 256, 256, 0, stream>>>(src, dst, ea, T1, ac1, dg, E);
  }
  node_relu_kernel<<<(((unsigned)N * 32u) + 255) / 256, 256, 0, stream>>>(
      ac1, dg, T1, b1, h, N);

  gemm_wmma_kernel<32, 192><<<gblocks, 256, 0, stream>>>(h, Wc2, T2, N);

  {
    unsigned total = (unsigned)E * 64u;
    edge_scatter_kernel<64, 192, false>
        <<<(total + 255) / 256, 256, 0, stream>>>(src, dst, ea, T2, ac2, dg, E);
  }
  node_logsoftmax_kernel<<<(((unsigned)N * 32u) + 255) / 256, 256, 0, stream>>>(
      ac2, dg, T2, b2, (float*)d_out, N);
}